// BilinearRoiPooling_90494960927361
// MI455X (gfx1250) — compile-verified
//
#include <hip/hip_runtime.h>
#include <cstdint>

#ifndef __has_builtin
#define __has_builtin(x) 0
#endif

#if __has_builtin(__builtin_amdgcn_tensor_load_to_lds)
#define HAVE_TDM 1
#else
#define HAVE_TDM 0
#endif

typedef unsigned int u32x4 __attribute__((ext_vector_type(4)));
typedef int          i32x8 __attribute__((ext_vector_type(8)));
typedef int          i32x4 __attribute__((ext_vector_type(4)));

#define C_TOT 512
#define HF    64
#define WF    256
#define HH_   7
#define WW_   7
#define CH    128              // channels staged per TDM chunk
#define NCHUNK (C_TOT / CH)
#define SPANMAX 36             // max x-span: 6*(256*255/(2047*6)) + 2 = 33.9 -> 36 w/ margin

// One workgroup per (box b, output row i).
// Stage feats[c0:c0+CH, y0:y0+2, xlo:xlo+S] -> LDS via the Tensor Data Mover,
// then bilinear-blend 128x7 outputs per chunk out of LDS.
__global__ __launch_bounds__(256)
void BilinearRoiPooling_tdm_kernel(const float* __restrict__ feats,
                                   const float* __restrict__ boxes,
                                   const int*   __restrict__ imgh_p,
                                   const int*   __restrict__ imgw_p,
                                   float*       __restrict__ out)
{
    __shared__ float patch[CH * 2 * SPANMAX];   // 36 KB

    const int blk = blockIdx.x;
    const int b   = blk / HH_;
    const int i   = blk - b * HH_;
    const int tid = threadIdx.x;

    const float imgw = (float)imgw_p[0];
    const float imgh = (float)imgh_p[0];
    const float invw = 1.0f / (imgw - 1.0f);
    const float invh = 1.0f / (imgh - 1.0f);

    const float xc = boxes[(size_t)b * 4 + 0];
    const float yc = boxes[(size_t)b * 4 + 1];
    const float bw = boxes[(size_t)b * 4 + 2];
    const float bh = boxes[(size_t)b * 4 + 3];

    // ---- vertical sample position for row i (uniform across WG) ----
    const float ty  = (float)(i - 3) * (1.0f / 3.0f);
    const float gy  = (2.0f * yc - (imgh - 1.0f)) * invh + (bh * invh) * ty;
    const float py  = (gy + 1.0f) * 0.5f * (float)(HF - 1);
    const float y0f = floorf(py);
    const int   y0  = (int)y0f;
    float wy1 = py - y0f;
    float wy0 = 1.0f - wy1;
    wy0 = (y0     >= 0 && y0     <= HF - 1) ? wy0 : 0.0f;   // zero-padding semantics
    wy1 = (y0 + 1 >= 0 && y0 + 1 <= HF - 1) ? wy1 : 0.0f;
    int yorig = y0 < 0 ? 0 : (y0 > HF - 2 ? HF - 2 : y0);   // in-bounds 2-row tile origin
    int ry0 = y0 - yorig;     ry0 = ry0 < 0 ? 0 : (ry0 > 1 ? 1 : ry0);
    int ry1 = y0 + 1 - yorig; ry1 = ry1 < 0 ? 0 : (ry1 > 1 ? 1 : ry1);

    // ---- horizontal span (uniform across WG; shared by all 7 j's) ----
    const float gxc = (2.0f * xc - (imgw - 1.0f)) * invw;
    const float gxs = bw * invw;
    const float px0 = ((gxc + gxs * -1.0f) + 1.0f) * 0.5f * (float)(WF - 1); // j=0
    const float px6 = ((gxc + gxs *  1.0f) + 1.0f) * 0.5f * (float)(WF - 1); // j=6
    const int x_first = (int)floorf(px0);
    const int x_last  = (int)floorf(px6) + 1;
    int S = x_last - x_first + 1;
    S = S < 2 ? 2 : (S > SPANMAX ? SPANMAX : S);
    int xlo = x_first;
    xlo = xlo < 0 ? 0 : (xlo > WF - S ? WF - S : xlo);      // in-bounds x tile origin

    // Keep an LDS store visible to the compiler (TDM writes are invisible in IR).
    if (imgw_p == nullptr) patch[tid] = 0.0f;

    for (int chunk = 0; chunk < NCHUNK; ++chunk) {
        const int c0 = chunk * CH;

#if HAVE_TDM
        if (tid == 0) {
            // ---- Tensor DMA descriptor (CDNA5 ISA) ----
            const unsigned lds_off = (unsigned)(unsigned long long)(void*)&patch[0];
            const unsigned long long ga = (unsigned long long)(const void*)
                (feats + ((size_t)c0 * (HF * WF) + (size_t)yorig * WF + (size_t)xlo));
            // Group 0: count=1 | lds_addr | global_addr[56:0] | type=2
            u32x4 g0 = { 1u,
                         lds_off,
                         (unsigned)ga,
                         (unsigned)((ga >> 32) & 0x01FFFFFFu) | 0x80000000u };
            // Group 1: data_size=4B; tensor dims (Wf,Hf); tile (S,2,CH);
            //          strides: dim0=Wf, dim1=Hf*Wf (elements)
            i32x8 g1 = { (int)(2u << 16),                         // data_size=2 (4B), mask=0
                         (int)((WF & 0xFFFF) << 16),              // tensor_dim0 lo
                         (int)(((unsigned)WF >> 16) | ((HF & 0xFFFF) << 16)), // dim0 hi | dim1 lo
                         (int)(((unsigned)HF >> 16) | ((unsigned)S << 16)),   // dim1 hi | tile_dim0
                         (int)(2u | ((unsigned)CH << 16)),        // tile_dim1=2 | tile_dim2=CH
                         (int)WF,                                 // tensor_dim0_stride lo
                         (int)(((HF * WF) & 0xFFFF) << 16),       // stride0 hi(0) | stride1 lo
                         (int)((unsigned)(HF * WF) >> 16) };      // stride1 hi
            i32x4 g2 = { C_TOT, 0, 0, 0 };                        // tensor_dim2=C
            i32x4 g3 = { 0, 0, 0, 0 };
            i32x8 g4 = { 0, 0, 0, 0, 0, 0, 0, 0 };               // unused extension words
            __builtin_amdgcn_tensor_load_to_lds(g0, g1, g2, g3, g4, 0);
#if __has_builtin(__builtin_amdgcn_s_wait_tensorcnt)
            __builtin_amdgcn_s_wait_tensorcnt(0);
#else
            asm volatile("s_wait_tensorcnt 0x0" ::: "memory");
#endif
        }
#else
        // Fallback: cooperative coalesced staging (identical layout).
        for (int idx = tid; idx < CH * 2 * S; idx += 256) {
            const int c   = idx / (2 * S);
            const int rem = idx - c * (2 * S);
            const int r   = rem / S;
            const int x   = rem - r * S;
            patch[idx] = feats[((size_t)(c0 + c) * HF + (size_t)(yorig + r)) * WF
                               + (size_t)(xlo + x)];
        }
#endif
        __syncthreads();

        // ---- blend: CH channels x 7 horizontal samples ----
        for (int e = tid; e < CH * WW_; e += 256) {
            const int c = e / WW_;
            const int j = e - c * WW_;

            const float tx  = (float)(j - 3) * (1.0f / 3.0f);
            const float gx  = gxc + gxs * tx;
            const float px  = (gx + 1.0f) * 0.5f * (float)(WF - 1);
            const float x0f = floorf(px);
            const int   x0  = (int)x0f;
            float wx1 = px - x0f;
            float wx0 = 1.0f - wx1;
            wx0 = (x0     >= 0 && x0     <= WF - 1) ? wx0 : 0.0f;
            wx1 = (x0 + 1 >= 0 && x0 + 1 <= WF - 1) ? wx1 : 0.0f;
            int rx0 = x0 - xlo;     rx0 = rx0 < 0 ? 0 : (rx0 > S - 1 ? S - 1 : rx0);
            int rx1 = x0 + 1 - xlo; rx1 = rx1 < 0 ? 0 : (rx1 > S - 1 ? S - 1 : rx1);

            const float* row0 = &patch[(c * 2 + ry0) * S];
            const float* row1 = &patch[(c * 2 + ry1) * S];
            float acc =      row0[rx0] * (wx0 * wy0);
            acc = fmaf(row0[rx1], wx1 * wy0, acc);
            acc = fmaf(row1[rx0], wx0 * wy1, acc);
            acc = fmaf(row1[rx1], wx1 * wy1, acc);

            out[(((size_t)b * C_TOT + (size_t)(c0 + c)) * HH_ + i) * WW_ + j] = acc;
        }
        __syncthreads();   // protect LDS before next chunk's DMA
    }
}

extern "C" void kernel_launch(void* const* d_in, const int* in_sizes, int n_in,
                              void* d_out, int out_size, void* d_ws, size_t ws_size,
                              hipStream_t stream) {
    (void)n_in; (void)out_size; (void)d_ws; (void)ws_size;
    const float* feats = (const float*)d_in[0];
    const float* boxes = (const float*)d_in[1];
    const int*   imgh  = (const int*)d_in[2];
    const int*   imgw  = (const int*)d_in[3];
    float*       out   = (float*)d_out;

    const int B = in_sizes[1] / 4;          // boxes: (B,4)
    dim3 grid((unsigned)(B * HH_));
    BilinearRoiPooling_tdm_kernel<<<grid, 256, 0, stream>>>(feats, boxes, imgh, imgw, out);
}